// HashEncoder_61237643706936
// MI455X (gfx1250) — compile-verified
//
#include <hip/hip_runtime.h>
#include <stdint.h>

typedef __attribute__((ext_vector_type(2))) float        v2f;
typedef __attribute__((ext_vector_type(4))) unsigned int v4u;
typedef __attribute__((ext_vector_type(4))) int          v4i;
typedef __attribute__((ext_vector_type(8))) int          v8i;

#define N_LEVELS      16
#define LOG2_T        19
#define TBL_SIZE      (1u << LOG2_T)
#define TBL_MASK      (TBL_SIZE - 1u)
#define N_POINTS      (1u << 20)
#define PTS_PER_BLOCK 16
#define BLOCK_SIZE    256
#define PRIME_Y       2654435761u
#define PRIME_Z       805459861u

// floor(16 * (2048/16)^(l/15)) for l = 0..15  (matches reference float64 math)
__constant__ float c_res[N_LEVELS] = {
    16.f,  22.f,  30.f,  42.f,  58.f,  80.f,  111.f,  153.f,
    212.f, 294.f, 406.f, 561.f, 776.f, 1072.f, 1482.f, 2048.f};

__global__ __launch_bounds__(BLOCK_SIZE) void HashEncoder_61237643706936_kernel(
    const float* __restrict__ x,      // [N_POINTS, 3]
    const float* __restrict__ table,  // [16, 2^19, 2]
    float* __restrict__ out)          // [N_POINTS, 32]
{
    __shared__ float sx[PTS_PER_BLOCK * 3];  // 16 points * xyz = 48 floats

    // ---- TDM: async-stage this block's 16 points (192 B) into LDS ----
    // Only wave 0 issues the tensor op (TDM ignores EXEC; one op per block).
    if (threadIdx.x < 32u) {
        uint64_t ga = (uint64_t)(uintptr_t)(x + (size_t)blockIdx.x * (PTS_PER_BLOCK * 3));
        // Generic LDS address: low 32 bits are the LDS byte offset (ISA 10.2).
        uint32_t la = (uint32_t)(uintptr_t)(&sx[0]);

        v4u g0;
        g0.x = 1u;                                               // count=1, user desc
        g0.y = la;                                               // lds_addr
        g0.z = (uint32_t)ga;                                     // global_addr[31:0]
        g0.w = ((uint32_t)(ga >> 32) & 0x01FFFFFFu) | (2u << 30); // addr[56:32] | type=2

        v8i g1;
        g1[0] = (int)(2u << 16);   // workgroup_mask=0 (no cluster), data_size=4 B
        g1[1] = (int)(48u << 16);  // tensor_dim0 = 48 elements ([79:48] -> w1[31:16])
        g1[2] = 0;                 // tensor_dim0 hi / tensor_dim1 = 0
        g1[3] = (int)(48u << 16);  // tile_dim0 = 48 ([127:112])
        g1[4] = 0;                 // tile_dim1 = tile_dim2 = 0 (1-D tile)
        g1[5] = 48;                // tensor_dim0_stride = 48
        g1[6] = 0;
        g1[7] = 0;

        v4i z4 = {0, 0, 0, 0};              // groups 2/3 unused (<=2-D tensor)
        v8i z8 = {0, 0, 0, 0, 0, 0, 0, 0};  // trailing zero group (6-arg form)
        __builtin_amdgcn_tensor_load_to_lds(g0, g1, z4, z4, z8, /*cpol=*/0);
        __builtin_amdgcn_s_wait_tensorcnt(0);
    }
    __syncthreads();

    // ---- per-(point, level) work: lane = (local_point<<4) | level ----
    const unsigned gid = blockIdx.x * BLOCK_SIZE + threadIdx.x;
    const unsigned p   = gid >> 4;            // global point id
    const unsigned l   = gid & 15u;           // level id
    const unsigned pl  = threadIdx.x >> 4;    // point id within block

    const float X = sx[pl * 3 + 0];           // LDS broadcast (16 lanes share)
    const float Y = sx[pl * 3 + 1];
    const float Z = sx[pl * 3 + 2];

    const float res = c_res[l];
    const float px = X * res, py = Y * res, pz = Z * res;
    const float bx = floorf(px), by = floorf(py), bz = floorf(pz);
    const float fx = px - bx, fy = py - by, fz = pz - bz;

    const unsigned ux = (unsigned)(int)bx;
    const unsigned uy = (unsigned)(int)by;
    const unsigned uz = (unsigned)(int)bz;

    // hash terms: PRIMES = {1, 2654435761, 805459861}
    const unsigned hx0 = ux,                  hx1 = ux + 1u;
    const unsigned hy0 = uy * PRIME_Y,        hy1 = (uy + 1u) * PRIME_Y;
    const unsigned hz0 = uz * PRIME_Z,        hz1 = (uz + 1u) * PRIME_Z;

    // 8 x float2 gathers from the L2-resident table (64 MB < 192 MB L2)
    const v2f* tbl = (const v2f*)table + (size_t)l * TBL_SIZE;
    const v2f f000 = tbl[(hx0 ^ hy0 ^ hz0) & TBL_MASK];
    const v2f f001 = tbl[(hx0 ^ hy0 ^ hz1) & TBL_MASK];
    const v2f f010 = tbl[(hx0 ^ hy1 ^ hz0) & TBL_MASK];
    const v2f f011 = tbl[(hx0 ^ hy1 ^ hz1) & TBL_MASK];
    const v2f f100 = tbl[(hx1 ^ hy0 ^ hz0) & TBL_MASK];
    const v2f f101 = tbl[(hx1 ^ hy0 ^ hz1) & TBL_MASK];
    const v2f f110 = tbl[(hx1 ^ hy1 ^ hz0) & TBL_MASK];
    const v2f f111 = tbl[(hx1 ^ hy1 ^ hz1) & TBL_MASK];

    // trilinear weights
    const float wx0 = 1.f - fx, wx1 = fx;
    const float wy0 = 1.f - fy, wy1 = fy;
    const float wz0 = 1.f - fz, wz1 = fz;
    const float w00 = wx0 * wy0, w01 = wx0 * wy1;
    const float w10 = wx1 * wy0, w11 = wx1 * wy1;

    v2f acc = f000 * (w00 * wz0);
    acc += f001 * (w00 * wz1);
    acc += f010 * (w01 * wz0);
    acc += f011 * (w01 * wz1);
    acc += f100 * (w10 * wz0);
    acc += f101 * (w10 * wz1);
    acc += f110 * (w11 * wz0);
    acc += f111 * (w11 * wz1);

    // out[p][l*2 .. l*2+1]; wave stores 256 contiguous bytes.
    // Non-temporal: 128 MB write-once stream must not evict the table from L2.
    __builtin_nontemporal_store(acc, (v2f*)(out + (size_t)p * (N_LEVELS * 2) + l * 2u));
}

extern "C" void kernel_launch(void* const* d_in, const int* in_sizes, int n_in,
                              void* d_out, int out_size, void* d_ws, size_t ws_size,
                              hipStream_t stream) {
    (void)in_sizes; (void)n_in; (void)d_ws; (void)ws_size; (void)out_size;
    const float* x     = (const float*)d_in[0];
    const float* table = (const float*)d_in[1];
    float* out         = (float*)d_out;

    const unsigned total  = N_POINTS * N_LEVELS;  // 16,777,216 threads
    const unsigned blocks = total / BLOCK_SIZE;   // 65,536 blocks (exact)
    hipLaunchKernelGGL(HashEncoder_61237643706936_kernel,
                       dim3(blocks), dim3(BLOCK_SIZE), 0, stream, x, table, out);
}